// TreatmentGNN_14525579395682
// MI455X (gfx1250) — compile-verified
//
#include <hip/hip_runtime.h>
#include <hip/hip_bf16.h>

typedef float v2f __attribute__((ext_vector_type(2)));
typedef float v8f __attribute__((ext_vector_type(8)));

#define XS 260   // LDS row stride for 256-wide X tile (pad 4 -> conflict-free)
#define HS 132   // LDS row stride for 128-wide H tile

// ---------------------------------------------------------------- zero scratch
__global__ __launch_bounds__(256)
void gnn_zero_kernel(float* __restrict__ p, size_t n) {
    size_t i = (size_t)blockIdx.x * blockDim.x + threadIdx.x;
    size_t stride = (size_t)gridDim.x * blockDim.x;
    for (; i < n; i += stride) p[i] = 0.0f;
}

// ------------------------------------------------- weighted gather/scatter-add
// one 32-lane group per edge; each lane moves a float4 (512B/wave gather).
// Edge arrays are streamed once -> non-temporal loads keep L2 for msg/deg/emb.
__global__ __launch_bounds__(256)
void gnn_scatter_kernel(const int* __restrict__ src, const int* __restrict__ dst,
                        const float* __restrict__ w,
                        const float* __restrict__ src_emb,
                        float* __restrict__ msg, float* __restrict__ deg, int nE) {
    long long tid = (long long)blockIdx.x * blockDim.x + threadIdx.x;
    int e = (int)(tid >> 5);
    int q = (int)(tid & 31);
    if (e >= nE) return;
    float we = __builtin_nontemporal_load(w + e);
    int s = __builtin_nontemporal_load(src + e);
    int d = __builtin_nontemporal_load(dst + e);
    float4 v = ((const float4*)(src_emb + (size_t)s * 128))[q];
    float* p = msg + (size_t)d * 128 + q * 4;
    __hip_atomic_fetch_add(p + 0, v.x * we, __ATOMIC_RELAXED, __HIP_MEMORY_SCOPE_AGENT);
    __hip_atomic_fetch_add(p + 1, v.y * we, __ATOMIC_RELAXED, __HIP_MEMORY_SCOPE_AGENT);
    __hip_atomic_fetch_add(p + 2, v.z * we, __ATOMIC_RELAXED, __HIP_MEMORY_SCOPE_AGENT);
    __hip_atomic_fetch_add(p + 3, v.w * we, __ATOMIC_RELAXED, __HIP_MEMORY_SCOPE_AGENT);
    if (q == 0)
        __hip_atomic_fetch_add(deg + d, we, __ATOMIC_RELAXED, __HIP_MEMORY_SCOPE_AGENT);
}

// ------------------------------------------------------------------ CDNA5 async
// GLOBAL_LOAD_ASYNC_TO_LDS_B128 (GV mode): per-lane 16B memory->LDS DMA,
// tracked with ASYNCcnt (ISA 15.18.3 op 98).
__device__ __forceinline__ void async_b128_to_lds(unsigned lds_off, const void* gaddr) {
    asm volatile("global_load_async_to_lds_b128 %0, %1, off"
                 :: "v"(lds_off), "v"(gaddr) : "memory");
}
__device__ __forceinline__ void wait_async0() {
    asm volatile("s_wait_asynccnt 0" ::: "memory");
}

// ------------------------------------------------------------------ wave helpers
__device__ __forceinline__ float wave_sum(float v) {
    #pragma unroll
    for (int off = 16; off > 0; off >>= 1) v += __shfl_xor(v, off, 32);
    return v;
}

// -------------------------------------------------------------------- fused MLP
// One wave per 16-node tile. X = [emb | msg/(deg+eps)] staged in LDS via
// async global->LDS copies. Layer1: (16x256)@(256x128) via
// V_WMMA_F32_16X16X4_F32, +b1, LN(g1,be1), ReLU. Layer2: (16x128)@(128x64),
// +b2, LN(g2,be2) -> out.
__global__ __launch_bounds__(32)
void gnn_mlp_kernel(const float* __restrict__ emb,
                    const float* __restrict__ msg,
                    const float* __restrict__ deg,
                    const float* __restrict__ W1, const float* __restrict__ b1,
                    const float* __restrict__ g1, const float* __restrict__ be1,
                    const float* __restrict__ W2, const float* __restrict__ b2,
                    const float* __restrict__ g2, const float* __restrict__ be2,
                    float* __restrict__ out) {
    __shared__ float X[16 * XS];
    __shared__ float Hh[16 * HS];

    const int lane = threadIdx.x;            // 0..31
    const int row0 = blockIdx.x * 16;
    const int nl   = lane & 15;              // fragment row/col index
    const int koff = (lane >> 4) << 1;       // 0 or 2 (K sub-offset per half-wave)
    const int mo   = (lane >> 4) << 3;       // 0 or 8 (M offset for C/D rows)

    // ---- stage X tile with async DMA: [emb | raw msg], 16B per lane per row
    const unsigned xbase = (unsigned)(size_t)&X[0];
    for (int r = 0; r < 16; ++r) {
        int node = row0 + r;
        async_b128_to_lds(xbase + (unsigned)(r * XS + 4 * lane) * 4u,
                          emb + (size_t)node * 128 + 4 * lane);
        async_b128_to_lds(xbase + (unsigned)(r * XS + 128 + 4 * lane) * 4u,
                          msg + (size_t)node * 128 + 4 * lane);
    }
    wait_async0();
    __syncthreads();

    // ---- normalize msg half in LDS: msg /= (deg + eps)
    for (int r = 0; r < 16; ++r) {
        float invd = 1.0f / (deg[row0 + r] + 1e-8f);
        float4 m4 = *(float4*)&X[r * XS + 128 + 4 * lane];
        m4.x *= invd; m4.y *= invd; m4.z *= invd; m4.w *= invd;
        *(float4*)&X[r * XS + 128 + 4 * lane] = m4;
    }
    __syncthreads();

    // ---- layer 1: H = X @ W1^T + b1  (8 N-tiles x 64 K-steps of 16x16x4 f32 WMMA)
    for (int nt = 0; nt < 8; ++nt) {
        const int N0 = nt * 16;
        v8f acc = {0.f, 0.f, 0.f, 0.f, 0.f, 0.f, 0.f, 0.f};
        const float* wrow = W1 + (size_t)(N0 + nl) * 256 + koff;
        const float* xrow = &X[nl * XS + koff];
        for (int k0 = 0; k0 < 256; k0 += 4) {
            v2f a; a[0] = xrow[k0]; a[1] = xrow[k0 + 1];
            v2f b; b[0] = wrow[k0]; b[1] = wrow[k0 + 1];
            acc = __builtin_amdgcn_wmma_f32_16x16x4_f32(false, a, false, b,
                                                        (short)0, acc, false, false);
        }
        float bias = b1[N0 + nl];
        #pragma unroll
        for (int v = 0; v < 8; ++v)
            Hh[(v + mo) * HS + N0 + nl] = acc[v] + bias;
    }
    __syncthreads();

    // ---- layernorm over 128 + ReLU, in place in Hh
    for (int r = 0; r < 16; ++r) {
        float x0 = Hh[r * HS + lane];
        float x1 = Hh[r * HS + lane + 32];
        float x2 = Hh[r * HS + lane + 64];
        float x3 = Hh[r * HS + lane + 96];
        float mu = wave_sum(x0 + x1 + x2 + x3) * (1.0f / 128.0f);
        float d0 = x0 - mu, d1 = x1 - mu, d2 = x2 - mu, d3 = x3 - mu;
        float var = wave_sum(d0 * d0 + d1 * d1 + d2 * d2 + d3 * d3) * (1.0f / 128.0f);
        float rinv = rsqrtf(var + 1e-5f);
        Hh[r * HS + lane]      = fmaxf(d0 * rinv * g1[lane]      + be1[lane],      0.f);
        Hh[r * HS + lane + 32] = fmaxf(d1 * rinv * g1[lane + 32] + be1[lane + 32], 0.f);
        Hh[r * HS + lane + 64] = fmaxf(d2 * rinv * g1[lane + 64] + be1[lane + 64], 0.f);
        Hh[r * HS + lane + 96] = fmaxf(d3 * rinv * g1[lane + 96] + be1[lane + 96], 0.f);
    }
    __syncthreads();

    // ---- layer 2: H2 = Hrelu @ W2^T + b2 (4 N-tiles x 32 K-steps), into X buffer
    for (int nt = 0; nt < 4; ++nt) {
        const int N0 = nt * 16;
        v8f acc = {0.f, 0.f, 0.f, 0.f, 0.f, 0.f, 0.f, 0.f};
        const float* wrow = W2 + (size_t)(N0 + nl) * 128 + koff;
        const float* hrow = &Hh[nl * HS + koff];
        for (int k0 = 0; k0 < 128; k0 += 4) {
            v2f a; a[0] = hrow[k0]; a[1] = hrow[k0 + 1];
            v2f b; b[0] = wrow[k0]; b[1] = wrow[k0 + 1];
            acc = __builtin_amdgcn_wmma_f32_16x16x4_f32(false, a, false, b,
                                                        (short)0, acc, false, false);
        }
        float bias = b2[N0 + nl];
        #pragma unroll
        for (int v = 0; v < 8; ++v)
            X[(v + mo) * XS + N0 + nl] = acc[v] + bias;
    }
    __syncthreads();

    // ---- layernorm over 64 + store
    for (int r = 0; r < 16; ++r) {
        float x0 = X[r * XS + lane];
        float x1 = X[r * XS + lane + 32];
        float mu = wave_sum(x0 + x1) * (1.0f / 64.0f);
        float d0 = x0 - mu, d1 = x1 - mu;
        float var = wave_sum(d0 * d0 + d1 * d1) * (1.0f / 64.0f);
        float rinv = rsqrtf(var + 1e-5f);
        float* orow = out + (size_t)(row0 + r) * 64;
        orow[lane]      = d0 * rinv * g2[lane]      + be2[lane];
        orow[lane + 32] = d1 * rinv * g2[lane + 32] + be2[lane + 32];
    }
}

// ------------------------------------------------------------------------ launch
extern "C" void kernel_launch(void* const* d_in, const int* in_sizes, int n_in,
                              void* d_out, int out_size, void* d_ws, size_t ws_size,
                              hipStream_t stream) {
    (void)n_in; (void)out_size; (void)ws_size;

    const int* pc_edges = (const int*)d_in[0];   // provider->code edges (2,E)
    const int* cp_edges = (const int*)d_in[1];   // code->provider edges (2,E)
    const float* w      = (const float*)d_in[2];
    const float* p_emb  = (const float*)d_in[3];
    const float* c_emb  = (const float*)d_in[4];
    const float* pW1  = (const float*)d_in[5];
    const float* pb1  = (const float*)d_in[6];
    const float* pg1  = (const float*)d_in[7];
    const float* pbe1 = (const float*)d_in[8];
    const float* pW2  = (const float*)d_in[9];
    const float* pb2  = (const float*)d_in[10];
    const float* pg2  = (const float*)d_in[11];
    const float* pbe2 = (const float*)d_in[12];
    const float* cW1  = (const float*)d_in[13];
    const float* cb1  = (const float*)d_in[14];
    const float* cg1  = (const float*)d_in[15];
    const float* cbe1 = (const float*)d_in[16];
    const float* cW2  = (const float*)d_in[17];
    const float* cb2  = (const float*)d_in[18];
    const float* cg2  = (const float*)d_in[19];
    const float* cbe2 = (const float*)d_in[20];

    const int E  = in_sizes[2];
    const int NP = in_sizes[3] / 128;
    const int NC = in_sizes[4] / 128;

    float* ws = (float*)d_ws;
    float* prov_msg = ws;
    float* prov_deg = prov_msg + (size_t)NP * 128;
    float* code_msg = prov_deg + NP;
    float* code_deg = code_msg + (size_t)NC * 128;
    size_t scratch_floats = (size_t)NP * 128 + NP + (size_t)NC * 128 + NC;

    float* out = (float*)d_out;

    gnn_zero_kernel<<<4096, 256, 0, stream>>>(ws, scratch_floats);

    long long sthreads = (long long)E * 32;
    int sblocks = (int)((sthreads + 255) / 256);
    // code -> provider messages (uses code_provider_edges)
    gnn_scatter_kernel<<<sblocks, 256, 0, stream>>>(cp_edges, cp_edges + E, w,
                                                    c_emb, prov_msg, prov_deg, E);
    // provider -> code messages (uses provider_code_edges)
    gnn_scatter_kernel<<<sblocks, 256, 0, stream>>>(pc_edges, pc_edges + E, w,
                                                    p_emb, code_msg, code_deg, E);

    gnn_mlp_kernel<<<NP / 16, 32, 0, stream>>>(p_emb, prov_msg, prov_deg,
                                               pW1, pb1, pg1, pbe1,
                                               pW2, pb2, pg2, pbe2,
                                               out);
    gnn_mlp_kernel<<<NC / 16, 32, 0, stream>>>(c_emb, code_msg, code_deg,
                                               cW1, cb1, cg1, cbe1,
                                               cW2, cb2, cg2, cbe2,
                                               out + (size_t)NP * 64);
}